// TransferNet_89395449299190
// MI455X (gfx1250) — compile-verified
//
#include <hip/hip_runtime.h>
#include <hip/hip_bf16.h>
#include <math.h>

// MI455X / gfx1250, wave32. f32 WMMA (V_WMMA_F32_16X16X4_F32) used for both GEMMs.
// Workspace layout (floats): xp[T*192] | hist[B*E*64] | ep0[B*E] | ep1[B*E] | le1[B*E] | cqw[512]
// Total ~184 MB.

typedef float v2f __attribute__((ext_vector_type(2)));
typedef float v8f __attribute__((ext_vector_type(8)));

#define B_ 4
#define E_ 100000
#define T_ 100000
#define D_ 64
#define G_ 192   // 3*D
#define EPS_ 1e-6f

__device__ __forceinline__ float sigmoidf_(float x) { return 1.0f / (1.0f + __expf(-x)); }

// ---------------------------------------------------------------------------
// cqw[t][b][d] = tanh(rel_emb[query[b]] @ W_step[t] + b_step[t])[d] * w_cls[d]
__global__ void k_cqw(const float* __restrict__ rel_emb, const float* __restrict__ W_step,
                      const float* __restrict__ b_step, const float* __restrict__ w_cls,
                      const int* __restrict__ query, float* __restrict__ cqw) {
    int tid = threadIdx.x;                 // 512 threads
    int t = tid >> 8;
    int b = (tid >> 6) & 3;
    int d = tid & 63;
    const float* q = rel_emb + (size_t)query[b] * D_;
    const float* W = W_step + (size_t)t * D_ * D_;
    float s = b_step[t * D_ + d];
    for (int k = 0; k < D_; ++k) s += q[k] * W[k * D_ + d];
    cqw[(t * B_ + b) * D_ + d] = tanhf(s) * w_cls[d];
}

// ---------------------------------------------------------------------------
// x_proj[i, 0:192] = rel_emb[rel[i]] @ w_ih^T + b_ih   (WMMA f32 16x16x4)
__global__ void __launch_bounds__(256) k_xproj(const float* __restrict__ rel_emb,
                                               const float* __restrict__ w_ih,
                                               const float* __restrict__ b_ih,
                                               const int* __restrict__ kb,
                                               float* __restrict__ xp) {
    __shared__ float wT[D_ * G_];          // wT[k*192 + g] = w_ih[g][k]  (48 KB)
    for (int idx = threadIdx.x; idx < D_ * G_; idx += 256) {
        int k = idx / G_, g = idx % G_;
        wT[idx] = w_ih[(size_t)g * D_ + k];
    }
    __syncthreads();

    int wave = threadIdx.x >> 5;
    int lane = threadIdx.x & 31;
    int tile = blockIdx.x * 8 + wave;      // 16-row tile index
    if (tile >= T_ / 16) return;           // wave-uniform guard (T % 16 == 0)
    int row0 = tile * 16;
    int m  = lane & 15;                    // A row / C column within tile
    int hi = lane >> 4;                    // K-half select

    int i = row0 + m;
    int reli = kb[3 * i + 2];
    const float* ap = rel_emb + (size_t)reli * D_;

    v2f a[16];                             // A fragments for all 16 K-steps
#pragma unroll
    for (int kk = 0; kk < 16; ++kk) {
        float2 t2 = *(const float2*)(ap + 4 * kk + 2 * hi);
        a[kk].x = t2.x; a[kk].y = t2.y;
    }

#pragma unroll
    for (int nt = 0; nt < 12; ++nt) {
        int col = nt * 16 + m;
        v8f acc = {};
#pragma unroll
        for (int kk = 0; kk < 16; ++kk) {
            int k0 = 4 * kk + 2 * hi;
            v2f bf;
            bf.x = wT[k0 * G_ + col];
            bf.y = wT[(k0 + 1) * G_ + col];
            acc = __builtin_amdgcn_wmma_f32_16x16x4_f32(
                false, a[kk], false, bf, (short)0, acc, false, false);
        }
        float bias = b_ih[nt * 16 + m];
#pragma unroll
        for (int v = 0; v < 8; ++v) {
            int r = v + 8 * hi;
            xp[(size_t)(row0 + r) * G_ + nt * 16 + m] = acc[v] + bias;
        }
    }
}

// ---------------------------------------------------------------------------
// Step 0: hist==0, so trans is batch-independent; scatter ep0 and hist numerator.
__global__ void __launch_bounds__(256) k_step0(const float* __restrict__ xp,
                                               const float* __restrict__ b_hh,
                                               const float* __restrict__ cqw,
                                               const float* __restrict__ start,
                                               const float* __restrict__ b_cls,
                                               const int* __restrict__ kb,
                                               float* __restrict__ ep0,
                                               float* __restrict__ histN) {
    __shared__ int sub_s[4], obj_s[4];
    __shared__ float part[4][2][4];        // [triple][wave-half][b]
    __shared__ float op_s[4][4];           // [triple][b]
    int tid = threadIdx.x;
    int tl = tid >> 6;                     // local triple 0..3
    int d  = tid & 63;
    if (tid < 4) {
        sub_s[tid] = kb[3 * (blockIdx.x * 4 + tid) + 0];
        obj_s[tid] = kb[3 * (blockIdx.x * 4 + tid) + 1];
    }
    __syncthreads();

    int i = blockIdx.x * 4 + tl;
    const float* xrow = xp + (size_t)i * G_;
    float r = sigmoidf_(xrow[d] + b_hh[d]);
    float z = sigmoidf_(xrow[64 + d] + b_hh[64 + d]);
    float n = tanhf(xrow[128 + d] + r * b_hh[128 + d]);
    float trans = (1.0f - z) * n;          // h == 0 at t=0

    float p0 = trans * cqw[0 * D_ + d];
    float p1 = trans * cqw[1 * D_ + d];
    float p2 = trans * cqw[2 * D_ + d];
    float p3 = trans * cqw[3 * D_ + d];
#pragma unroll
    for (int off = 16; off >= 1; off >>= 1) {
        p0 += __shfl_xor(p0, off);
        p1 += __shfl_xor(p1, off);
        p2 += __shfl_xor(p2, off);
        p3 += __shfl_xor(p3, off);
    }
    int wh = (tid >> 5) & 1;
    if ((tid & 31) == 0) {
        part[tl][wh][0] = p0; part[tl][wh][1] = p1;
        part[tl][wh][2] = p2; part[tl][wh][3] = p3;
    }
    __syncthreads();
    if (tid < 16) {
        int t2 = tid >> 2, b = tid & 3;
        float logit = part[t2][0][b] + part[t2][1][b] + b_cls[0];
        float tp = sigmoidf_(logit);
        float op = start[(size_t)b * E_ + sub_s[t2]] * tp;
        op_s[t2][b] = op;
        atomicAdd(&ep0[(size_t)b * E_ + obj_s[t2]], op);
    }
    __syncthreads();
    size_t base = (size_t)obj_s[tl] * D_ + d;
#pragma unroll
    for (int b = 0; b < 4; ++b)
        atomicAdd(&histN[(size_t)b * E_ * D_ + base], trans * op_s[tl][b]);
}

// ---------------------------------------------------------------------------
// hist = histN / (ep0 + eps);  le1 = (ep0 > 1) ? 1 : ep0
__global__ void k_norm(const float* __restrict__ ep0, float* __restrict__ histN,
                       float* __restrict__ le1) {
    int idx = blockIdx.x * blockDim.x + threadIdx.x;
    if (idx >= B_ * E_) return;
    float ep = ep0[idx];
    le1[idx] = (ep > 1.0f) ? 1.0f : ep;
    float inv = 1.0f / (ep + EPS_);
    float4* h = (float4*)(histN + (size_t)idx * D_);
#pragma unroll
    for (int q = 0; q < 16; ++q) {
        float4 v = h[q];
        v.x *= inv; v.y *= inv; v.z *= inv; v.w *= inv;
        h[q] = v;
    }
}

// ---------------------------------------------------------------------------
// Step 1 fused: gather h, WMMA h@w_hh^T (96 wmma/wave), gates, logit, ep1 scatter.
// Block = 16 triples, 8 waves: wave = (b, col-half); hist scatter skipped (unused).
__global__ void __launch_bounds__(256) k_step1(const float* __restrict__ xp,
                                               const float* __restrict__ hist,
                                               const float* __restrict__ le1,
                                               const float* __restrict__ w_hh,
                                               const float* __restrict__ b_hh,
                                               const float* __restrict__ cqw,
                                               const float* __restrict__ b_cls,
                                               const int* __restrict__ kb,
                                               float* __restrict__ ep1) {
    __shared__ float wT[D_ * G_];          // wT[k*192 + g] = w_hh[g][k]  (48 KB)
    __shared__ int sub_s[16], obj_s[16];
    __shared__ float logit_s[4][16];
    int tid = threadIdx.x;
    for (int idx = tid; idx < D_ * G_; idx += 256) {
        int k = idx / G_, g = idx % G_;
        wT[idx] = w_hh[(size_t)g * D_ + k];
    }
    int i0 = blockIdx.x * 16;
    if (tid < 16) {
        sub_s[tid] = kb[3 * (i0 + tid) + 0];
        obj_s[tid] = kb[3 * (i0 + tid) + 1];
    }
    if (tid < 64) logit_s[tid >> 4][tid & 15] = 0.0f;
    __syncthreads();

    int wave = tid >> 5, lane = tid & 31;
    int b    = wave >> 1;                  // batch
    int half = wave & 1;                   // column half (d in [half*32, half*32+32))
    int m  = lane & 15;
    int hi = lane >> 4;

    const float* hrow = hist + ((size_t)b * E_ + sub_s[m]) * D_;
    v2f a[16];
#pragma unroll
    for (int kk = 0; kk < 16; ++kk) {
        float2 t2 = *(const float2*)(hrow + 4 * kk + 2 * hi);
        a[kk].x = t2.x; a[kk].y = t2.y;
    }

    float plog[8];
#pragma unroll
    for (int v = 0; v < 8; ++v) plog[v] = 0.0f;
    const float* cq1 = cqw + (1 * B_ + b) * D_;

    for (int j = 0; j < 2; ++j) {
        int dcol = half * 32 + j * 16 + m; // column within a 64-wide gate block
        v8f ar = {}, az = {}, an = {};
#pragma unroll
        for (int kk = 0; kk < 16; ++kk) {
            int k0 = 4 * kk + 2 * hi;
            v2f br, bz, bn;
            br.x = wT[k0 * G_ + dcol];        br.y = wT[(k0 + 1) * G_ + dcol];
            bz.x = wT[k0 * G_ + 64 + dcol];   bz.y = wT[(k0 + 1) * G_ + 64 + dcol];
            bn.x = wT[k0 * G_ + 128 + dcol];  bn.y = wT[(k0 + 1) * G_ + 128 + dcol];
            ar = __builtin_amdgcn_wmma_f32_16x16x4_f32(false, a[kk], false, br, (short)0, ar, false, false);
            az = __builtin_amdgcn_wmma_f32_16x16x4_f32(false, a[kk], false, bz, (short)0, az, false, false);
            an = __builtin_amdgcn_wmma_f32_16x16x4_f32(false, a[kk], false, bn, (short)0, an, false, false);
        }
        float bhr = b_hh[dcol], bhz = b_hh[64 + dcol], bhn = b_hh[128 + dcol];
        float cqv = cq1[dcol];
#pragma unroll
        for (int v = 0; v < 8; ++v) {
            int row = v + 8 * hi;
            const float* xrow = xp + (size_t)(i0 + row) * G_;
            float hval = hist[((size_t)b * E_ + sub_s[row]) * D_ + dcol];
            float r = sigmoidf_(xrow[dcol] + ar[v] + bhr);
            float z = sigmoidf_(xrow[64 + dcol] + az[v] + bhz);
            float n = tanhf(xrow[128 + dcol] + r * (an[v] + bhn));
            float trans = (1.0f - z) * n + z * hval;
            plog[v] += trans * cqv;
        }
    }
    // reduce over the 16 columns held by each 16-lane half (C layout: lane = column)
#pragma unroll
    for (int v = 0; v < 8; ++v) {
        float p = plog[v];
        p += __shfl_xor(p, 1);
        p += __shfl_xor(p, 2);
        p += __shfl_xor(p, 4);
        p += __shfl_xor(p, 8);
        plog[v] = p;
    }
    if (m == 0) {
#pragma unroll
        for (int v = 0; v < 8; ++v)
            atomicAdd(&logit_s[b][v + 8 * hi], plog[v]);
    }
    __syncthreads();
    if (tid < 64) {
        int bb = tid >> 4, mm = tid & 15;
        float logit = logit_s[bb][mm] + b_cls[0];
        float tp = sigmoidf_(logit);
        float op = le1[(size_t)bb * E_ + sub_s[mm]] * tp;
        atomicAdd(&ep1[(size_t)bb * E_ + obj_s[mm]], op);
    }
}

// ---------------------------------------------------------------------------
__global__ void k_final(const float* __restrict__ ep1, float* __restrict__ out) {
    int idx = blockIdx.x * blockDim.x + threadIdx.x;
    if (idx >= B_ * E_) return;
    float ep = ep1[idx];
    out[idx] = (ep > 1.0f) ? 1.0f : ep;
}

// ---------------------------------------------------------------------------
extern "C" void kernel_launch(void* const* d_in, const int* in_sizes, int n_in,
                              void* d_out, int out_size, void* d_ws, size_t ws_size,
                              hipStream_t stream) {
    (void)in_sizes; (void)n_in; (void)out_size; (void)ws_size;
    const float* start   = (const float*)d_in[0];
    const float* rel_emb = (const float*)d_in[1];
    const float* W_step  = (const float*)d_in[2];
    const float* b_step  = (const float*)d_in[3];
    const float* w_ih    = (const float*)d_in[4];
    const float* w_hh    = (const float*)d_in[5];
    const float* b_ih    = (const float*)d_in[6];
    const float* b_hh    = (const float*)d_in[7];
    const float* w_cls   = (const float*)d_in[8];
    const float* b_cls   = (const float*)d_in[9];
    const int*   query   = (const int*)d_in[10];
    const int*   kb      = (const int*)d_in[11];

    float* ws   = (float*)d_ws;
    float* xp   = ws;                               // T*192
    float* hist = xp + (size_t)T_ * G_;             // B*E*64
    float* ep0  = hist + (size_t)B_ * E_ * D_;      // B*E
    float* ep1  = ep0 + (size_t)B_ * E_;            // B*E
    float* le1  = ep1 + (size_t)B_ * E_;            // B*E
    float* cqw  = le1 + (size_t)B_ * E_;            // 512

    hipMemsetAsync(hist, 0, (size_t)B_ * E_ * D_ * sizeof(float), stream);
    hipMemsetAsync(ep0, 0, (size_t)B_ * E_ * sizeof(float), stream);
    hipMemsetAsync(ep1, 0, (size_t)B_ * E_ * sizeof(float), stream);

    k_cqw  <<<1, 512, 0, stream>>>(rel_emb, W_step, b_step, w_cls, query, cqw);
    k_xproj<<<(T_ / 16 + 7) / 8, 256, 0, stream>>>(rel_emb, w_ih, b_ih, kb, xp);
    k_step0<<<T_ / 4, 256, 0, stream>>>(xp, b_hh, cqw, start, b_cls, kb, ep0, hist);
    k_norm <<<(B_ * E_ + 255) / 256, 256, 0, stream>>>(ep0, hist, le1);
    k_step1<<<T_ / 16, 256, 0, stream>>>(xp, hist, le1, w_hh, b_hh, cqw, b_cls, kb, ep1);
    k_final<<<(B_ * E_ + 255) / 256, 256, 0, stream>>>(ep1, (float*)d_out);
}